// AR_model_82824149336549
// MI455X (gfx1250) — compile-verified
//
#include <hip/hip_runtime.h>

// ---------------------------------------------------------------------------
// AR(64) over 65536 steps, batched broadcast-add of 256 x 65536 noise.
//
//   means block recurrence:  n_{j+1} = M n_j + v   (n_j = means[64j .. 64j+63])
//   M = (I - L)^-1 C  built from params by forward substitution.
//   Parallelized via matrix squaring (WMMA f32 16x16x4 GEMMs) + two-level
//   affine scan; final kernel is the pure-bandwidth broadcast (float4).
// ---------------------------------------------------------------------------

typedef __attribute__((ext_vector_type(2))) float v2f;
typedef __attribute__((ext_vector_type(8))) float v8f;

#define PORD   64      // AR order == block size
#define T_LEN  65536
#define BATCH  256
#define NCHUNK 32      // chunks (super-blocks)
#define CHUNK  32      // 64-blocks per chunk; NCHUNK*CHUNK*PORD == T_LEN

// workspace layout (float offsets)
#define WS_POW   0                       // pow[k] = M^(2^k), k=0..5, 4096 each
#define WS_U     (WS_POW + 6 * 4096)     // 64: u_k = (sum_{i<2^k} M^i) v
#define WS_V     (WS_U + 64)             // 64
#define WS_T     (WS_V + 64)             // 32*64: t_g = n_{32g}
#define WS_MEANS (WS_T + NCHUNK * PORD)  // 65536
// total = 92288 floats = 369,152 bytes of ws

// ---------------------------------------------------------------------------
// K1: build M (pow[0]) and v / u0 from params & bias.  One WG, 64 threads.
// Thread j owns column j of M.  M[r][j] = C[r][j] + sum_{c<r} a[r-1-c] M[c][j]
// with C[r][j] = (j >= r) ? a[63 - j + r] : 0.
// ---------------------------------------------------------------------------
__global__ void k_build(const float* __restrict__ params,
                        const float* __restrict__ bias,
                        float* __restrict__ ws) {
    __shared__ float a[PORD];
    const int j = threadIdx.x;                // 0..63
    a[j] = params[j];
    __syncthreads();

    float col[PORD];
    for (int r = 0; r < PORD; ++r) {
        float acc = (j >= r) ? a[63 - j + r] : 0.0f;
        for (int c = 0; c < r; ++c) acc = fmaf(a[r - 1 - c], col[c], acc);
        col[r] = acc;
    }
    float* M = ws + WS_POW;
    for (int r = 0; r < PORD; ++r) M[r * PORD + j] = col[r];

    if (j == 0) {
        const float b = bias[0];
        float v[PORD];
        for (int r = 0; r < PORD; ++r) {
            float acc = b;
            for (int c = 0; c < r; ++c) acc = fmaf(a[r - 1 - c], v[c], acc);
            v[r] = acc;
        }
        for (int r = 0; r < PORD; ++r) {
            ws[WS_V + r] = v[r];
            ws[WS_U + r] = v[r];              // u_0 = v
        }
    }
}

// ---------------------------------------------------------------------------
// K2: D = A * B for 64x64 f32 via V_WMMA_F32_16X16X4_F32.
// 512 threads = 16 waves; wave w handles C-tile (ti,tj) = (w/4, w%4),
// chaining K in 16 steps of 4.  Uniform control flow (EXEC all ones).
// ---------------------------------------------------------------------------
__global__ void k_gemm64(const float* __restrict__ A,
                         const float* __restrict__ B,
                         float* __restrict__ D) {
    const int lane = threadIdx.x & 31;
    const int wave = threadIdx.x >> 5;        // 0..15
    const int ti = wave >> 2, tj = wave & 3;
    const int half = lane >> 4;               // K sub-pair select
    const int mn = lane & 15;
    const int arow = ti * 16 + mn;            // A: M index for this lane
    const int bcol = tj * 16 + mn;            // B: N index for this lane

    v8f c = {};
    for (int kk = 0; kk < 16; ++kk) {
        const int k0 = kk * 4 + half * 2;
        v2f av, bv;
        av.x = A[arow * 64 + k0];             // A 16x4 tile: lane holds K=k0,k0+1
        av.y = A[arow * 64 + k0 + 1];
        bv.x = B[k0 * 64 + bcol];             // B 4x16 tile
        bv.y = B[(k0 + 1) * 64 + bcol];
        c = __builtin_amdgcn_wmma_f32_16x16x4_f32(
                /*neg_a=*/false, av, /*neg_b=*/false, bv,
                /*c_mod=*/(short)0, c, /*reuse_a=*/false, /*reuse_b=*/false);
    }
    // C/D layout: VGPR r -> M = r (lanes 0-15) or r+8 (lanes 16-31)
    for (int r = 0; r < 8; ++r) {
        const int row = ti * 16 + r + (half ? 8 : 0);
        D[row * 64 + bcol] = c[r];
    }
}

// ---------------------------------------------------------------------------
// K2b: u <- Pm * u + u   (doubling the geometric-sum vector).  64 threads.
// ---------------------------------------------------------------------------
__global__ void k_uvec(const float* __restrict__ Pm, float* __restrict__ u) {
    __shared__ float s[PORD];
    const int r = threadIdx.x;
    s[r] = u[r];
    __syncthreads();
    float acc = s[r];
    for (int k = 0; k < PORD; ++k) acc = fmaf(Pm[r * PORD + k], s[k], acc);
    u[r] = acc;
}

// ---------------------------------------------------------------------------
// K3: sequential super-chain  t_0 = v ; t_{g+1} = H t_g + w  (H = M^32).
// 64 threads; H-row in registers, state broadcast from LDS.
// ---------------------------------------------------------------------------
__global__ void k_tchain(const float* __restrict__ H, const float* __restrict__ w,
                         const float* __restrict__ v, float* __restrict__ t) {
    __shared__ float s[PORD];
    const int r = threadIdx.x;
    float hr[PORD];
    for (int k = 0; k < PORD; ++k) hr[k] = H[r * PORD + k];
    const float wr = w[r];
    float cur = v[r];
    s[r] = cur;
    t[r] = cur;
    __syncthreads();
    for (int g = 1; g < NCHUNK; ++g) {
        float acc = wr;
        for (int k = 0; k < PORD; ++k) acc = fmaf(hr[k], s[k], acc);
        __syncthreads();                      // all reads of s done
        s[r] = acc;
        t[g * PORD + r] = acc;
        __syncthreads();
    }
}

// ---------------------------------------------------------------------------
// K4: 32 parallel chunks; chunk g fills blocks 32g..32g+31 with n = M n + v.
// ---------------------------------------------------------------------------
__global__ void k_blocks(const float* __restrict__ M, const float* __restrict__ v,
                         const float* __restrict__ t, float* __restrict__ means) {
    __shared__ float s[PORD];
    const int r = threadIdx.x;
    const int g = blockIdx.x;
    float mr[PORD];
    for (int k = 0; k < PORD; ++k) mr[k] = M[r * PORD + k];
    const float vr = v[r];
    float cur = t[g * PORD + r];              // n_{32g}
    s[r] = cur;
    means[(g * CHUNK) * PORD + r] = cur;
    __syncthreads();
    for (int j = 1; j < CHUNK; ++j) {
        float acc = vr;
        for (int k = 0; k < PORD; ++k) acc = fmaf(mr[k], s[k], acc);
        __syncthreads();
        s[r] = acc;
        means[(g * CHUNK + j) * PORD + r] = acc;
        __syncthreads();
    }
}

// ---------------------------------------------------------------------------
// K5: out[b,t] = means[t] + 0.3*noise[b,t].  Pure bandwidth: float4 (b128)
// coalesced; grid.x covers T/4, grid.y covers batch.  means stays L2-hot.
// ---------------------------------------------------------------------------
__global__ void k_broadcast(const float* __restrict__ means,
                            const float* __restrict__ noise,
                            float* __restrict__ out) {
    const int t4 = blockIdx.x * blockDim.x + threadIdx.x;   // 0..16383
    const long i = (long)blockIdx.y * (T_LEN / 4) + t4;
    const float4 m = reinterpret_cast<const float4*>(means)[t4];
    const float4 n = reinterpret_cast<const float4*>(noise)[i];
    float4 o;
    o.x = fmaf(0.3f, n.x, m.x);
    o.y = fmaf(0.3f, n.y, m.y);
    o.z = fmaf(0.3f, n.z, m.z);
    o.w = fmaf(0.3f, n.w, m.w);
    reinterpret_cast<float4*>(out)[i] = o;
}

// ---------------------------------------------------------------------------
extern "C" void kernel_launch(void* const* d_in, const int* in_sizes, int n_in,
                              void* d_out, int out_size, void* d_ws, size_t ws_size,
                              hipStream_t stream) {
    const float* params = (const float*)d_in[0];   // [64]
    const float* bias   = (const float*)d_in[1];   // [1]
    const float* noise  = (const float*)d_in[2];   // [256*65536]
    float* out = (float*)d_out;
    float* ws  = (float*)d_ws;                      // needs ~369 KB

    // 1) Build M = (I-L)^-1 C, v, u0
    k_build<<<1, PORD, 0, stream>>>(params, bias, ws);

    // 2) Squaring chain M^2..M^32 (WMMA GEMMs) + geometric-sum vector doubling
    for (int k = 0; k < 5; ++k) {
        const float* Pk  = ws + WS_POW + k * 4096;
        float*       Pk1 = ws + WS_POW + (k + 1) * 4096;
        k_uvec<<<1, PORD, 0, stream>>>(Pk, ws + WS_U);     // u_{k+1} = Pk u_k + u_k
        k_gemm64<<<1, 512, 0, stream>>>(Pk, Pk, Pk1);      // M^(2^{k+1})
    }

    // 3) Super-chain of 32 states with H = M^32, w = u_5
    k_tchain<<<1, PORD, 0, stream>>>(ws + WS_POW + 5 * 4096, ws + WS_U,
                                     ws + WS_V, ws + WS_T);

    // 4) 32 parallel chunks fill all 1024 mean-blocks
    k_blocks<<<NCHUNK, PORD, 0, stream>>>(ws + WS_POW, ws + WS_V,
                                          ws + WS_T, ws + WS_MEANS);

    // 5) Bandwidth-bound broadcast add
    k_broadcast<<<dim3(T_LEN / 4 / 256, BATCH), 256, 0, stream>>>(
        ws + WS_MEANS, noise, out);
}